// RecurrentCRF_20426864460243
// MI455X (gfx1250) — compile-verified
//
#include <hip/hip_runtime.h>
#include <cstdint>
#include <cstddef>

// ---------------- problem constants (match reference) ----------------
#define NCLS   4
#define BATCH  16
#define HH     64
#define WW     512
#define K14    14
#define PLANE  (HH * WW)          // 32768 elements per [h,w] plane

// ---------------- tiling ----------------
#define TH 16
#define TW 64
#define TR (TH + 2)               // 18 rows incl. halo (pz = 1)
#define TC (TW + 4)               // 68 cols incl. halo (pa = 2)
#define NTHREADS 256              // 8 wave32

// Gaussian weights exp(-d2/(2*0.9^2)) for d2 in {1,2,4,5} (g_ang == g_bi)
#define G1 0.53940706f
#define G2 0.29096046f
#define G4 0.08465799f
#define G5 0.04566227f

// CDNA5 async global->LDS load (VDST = LDS byte address VGPR, VADDR = 64b global addr)
__device__ __forceinline__ void async_ld_b32(uint32_t lds_addr, const void* gaddr) {
    asm volatile("global_load_async_to_lds_b32 %0, %1, off"
                 :: "v"(lds_addr), "v"(gaddr)
                 : "memory");
}
__device__ __forceinline__ void wait_asynccnt0() {
    asm volatile("s_wait_asynccnt 0" ::: "memory");
}
__device__ __forceinline__ uint32_t lds_off(const void* p) {
    // flat LDS pointer: low 32 bits are the wave-relative LDS byte address
    return (uint32_t)(uintptr_t)p;
}

__global__ __launch_bounds__(NTHREADS)
void RecurrentCRF_step_kernel(const float* __restrict__ xin,
                              const float* __restrict__ bf,
                              const float* __restrict__ mask,
                              float* __restrict__ xout)
{
    // unary, unary*mask, raw-mask tiles (incl. halo). 9 * 18 * 68 * 4B = 44,064 B LDS.
    __shared__ float s_u [NCLS][TR][TC];
    __shared__ float s_um[NCLS][TR][TC];
    __shared__ float s_m [TR][TC];

    const int tid = threadIdx.x;
    const int w0  = blockIdx.x * TW;
    const int h0  = blockIdx.y * TH;
    const int b   = blockIdx.z;

    const float* xb  = xin  + (size_t)b * NCLS * PLANE;
    const float* mb  = mask + (size_t)b * PLANE;
    const float* bfb = bf   + (size_t)b * K14  * PLANE;
    float*       ob  = xout + (size_t)b * NCLS * PLANE;

    // ---- Stage 1: async-stage raw x (+halo) and mask into LDS (in-bounds only) ----
    for (int i = tid; i < TR * TC; i += NTHREADS) {
        const int r  = i / TC;
        const int c2 = i - r * TC;
        const int gh = h0 + r - 1;
        const int gw = w0 + c2 - 2;
        if ((unsigned)gh < (unsigned)HH && (unsigned)gw < (unsigned)WW) {
            const size_t g = (size_t)gh * WW + gw;
            #pragma unroll
            for (int c = 0; c < NCLS; ++c)
                async_ld_b32(lds_off(&s_u[c][r][c2]), xb + (size_t)c * PLANE + g);
            async_ld_b32(lds_off(&s_m[r][c2]), mb + g);
        }
    }
    wait_asynccnt0();
    __syncthreads();

    // ---- Stage 2: in-LDS softmax over classes; OOB tile cells -> 0 (pad AFTER softmax) ----
    for (int i = tid; i < TR * TC; i += NTHREADS) {
        const int r  = i / TC;
        const int c2 = i - r * TC;
        const int gh = h0 + r - 1;
        const int gw = w0 + c2 - 2;
        const bool inb = (unsigned)gh < (unsigned)HH && (unsigned)gw < (unsigned)WW;
        float u0 = 0.f, u1 = 0.f, u2 = 0.f, u3 = 0.f, m = 0.f;
        if (inb) {
            const float x0 = s_u[0][r][c2], x1 = s_u[1][r][c2];
            const float x2 = s_u[2][r][c2], x3 = s_u[3][r][c2];
            m = s_m[r][c2];
            const float mx  = fmaxf(fmaxf(x0, x1), fmaxf(x2, x3));
            const float e0 = __expf(x0 - mx), e1 = __expf(x1 - mx);
            const float e2 = __expf(x2 - mx), e3 = __expf(x3 - mx);
            const float inv = 1.0f / (e0 + e1 + e2 + e3);
            u0 = e0 * inv; u1 = e1 * inv; u2 = e2 * inv; u3 = e3 * inv;
        }
        s_u [0][r][c2] = u0;      s_u [1][r][c2] = u1;
        s_u [2][r][c2] = u2;      s_u [3][r][c2] = u3;
        s_um[0][r][c2] = u0 * m;  s_um[1][r][c2] = u1 * m;
        s_um[2][r][c2] = u2 * m;  s_um[3][r][c2] = u3 * m;
    }
    __syncthreads();

    // ---- Stage 3: 14-tap stencil + compat transform, core pixels only ----
    // neighbor tile coords for core pixel (r,c): (r + dz, c + da)
    const int   DZ[K14] = {0,0,0,0,0, 1,1,1,1, 2,2,2,2,2};
    const int   DA[K14] = {0,1,2,3,4, 0,1,3,4, 0,1,2,3,4};
    const float GW[K14] = {G5,G2,G1,G2,G5, G4,G1,G1,G4, G5,G2,G1,G2,G5};

    for (int p = tid; p < TH * TW; p += NTHREADS) {
        const int r  = p / TW;            // lanes walk stride-1 cols -> conflict-free banks
        const int cw = p - r * TW;
        const int rr = r + 1, cc = cw + 2;
        const int gh = h0 + r, gw = w0 + cw;
        const size_t g = (size_t)gh * WW + gw;

        float bfv[K14];
        #pragma unroll
        for (int k = 0; k < K14; ++k)
            bfv[k] = bfb[(size_t)k * PLANE + g];    // coalesced, reused by all 4 classes

        const float m = s_m[rr][cc];

        float u[NCLS], ang[NCLS], bi[NCLS];
        float Sa = 0.f, Sb = 0.f;
        #pragma unroll
        for (int c = 0; c < NCLS; ++c) {
            float a = 0.f, cond = 0.f;
            #pragma unroll
            for (int k = 0; k < K14; ++k) {
                const int nr = r  + DZ[k];
                const int nc = cw + DA[k];
                a    = fmaf(GW[k],  s_u [c][nr][nc], a);
                cond = fmaf(bfv[k], s_um[c][nr][nc], cond);
            }
            u[c]   = s_u[c][rr][cc];
            ang[c] = a;                   // g_ang == g_bi  =>  bi_ang == ang
            bi[c]  = cond * m * a;
            Sa += a;
            Sb += bi[c];
        }
        #pragma unroll
        for (int c = 0; c < NCLS; ++c) {
            // compat = coef * (1 - I):  out_o = coef * (sum_c v_c - v_o)
            ob[(size_t)c * PLANE + g] =
                u[c] + 0.02f * (Sa - ang[c]) + 0.1f * (Sb - bi[c]);
        }
    }
}

extern "C" void kernel_launch(void* const* d_in, const int* in_sizes, int n_in,
                              void* d_out, int out_size, void* d_ws, size_t ws_size,
                              hipStream_t stream)
{
    (void)in_sizes; (void)n_in; (void)out_size;
    const float* x    = (const float*)d_in[0];   // [B,C,H,W]
    const float* bf   = (const float*)d_in[1];   // [B,1,14,H,W]
    const float* mask = (const float*)d_in[2];   // [B,1,H,W]
    float* out = (float*)d_out;
    float* tmp = (float*)d_ws;                   // needs 8 MB; buffers stay L2-resident
    (void)ws_size;

    dim3 grid(WW / TW, HH / TH, BATCH);          // 8 x 4 x 16 = 512 workgroups
    dim3 block(NTHREADS);

    // iter1: x -> out ; iter2: out -> tmp ; iter3: tmp -> out   (ping-pong via L2)
    RecurrentCRF_step_kernel<<<grid, block, 0, stream>>>(x,   bf, mask, out);
    RecurrentCRF_step_kernel<<<grid, block, 0, stream>>>(out, bf, mask, tmp);
    RecurrentCRF_step_kernel<<<grid, block, 0, stream>>>(tmp, bf, mask, out);
}